// HybridModel_85615878078998
// MI455X (gfx1250) — compile-verified
//
#include <hip/hip_runtime.h>
#include <cstdint>
#include <cstddef>

// Channels are fixed by the model: IN = H = 128, OUT = 16.
#define C128 128
#define C16  16

typedef float v2f __attribute__((ext_vector_type(2)));
typedef float v8f __attribute__((ext_vector_type(8)));

// ---------------------------------------------------------------- utilities

__global__ void __launch_bounds__(256)
zero_kernel(float* __restrict__ p, size_t n) {
  const size_t n4 = n >> 2;
  float4* p4 = (float4*)p;
  const float4 z = {0.f, 0.f, 0.f, 0.f};
  size_t i = (size_t)blockIdx.x * blockDim.x + threadIdx.x;
  const size_t stride = (size_t)gridDim.x * blockDim.x;
  for (; i < n4; i += stride) p4[i] = z;
  if (blockIdx.x == 0 && threadIdx.x == 0)
    for (size_t j = n4 << 2; j < n; ++j) p[j] = 0.0f;
}

// ------------------------------------------------------- edge scatter (SAGE)
// One wave32 per edge; each lane handles 4 of the 128 channels.
// agg[dst] += x[src]; cnt[dst] += 1 (only on the first pass, cnt != nullptr).
__global__ void __launch_bounds__(256)
scatter_sum_kernel(const float* __restrict__ x, const int* __restrict__ src,
                   const int* __restrict__ dst, float* __restrict__ agg,
                   float* __restrict__ cnt, int E) {
  int e = blockIdx.x * 8 + (threadIdx.x >> 5);
  if (e >= E) return;
  int lane = threadIdx.x & 31;
  int s = src[e], d = dst[e];
  const float4 v = *(const float4*)(x + (size_t)s * C128 + lane * 4);
  float* ap = agg + (size_t)d * C128 + lane * 4;
  atomicAdd(ap + 0, v.x);
  atomicAdd(ap + 1, v.y);
  atomicAdd(ap + 2, v.z);
  atomicAdd(ap + 3, v.w);
  if (cnt != nullptr && lane == 0) atomicAdd(cnt + d, 1.0f);
}

// -------------------------------------------------------- edge scatter (GCN)
// agg[dst] += h[src] * rsqrt(deg_src) * rsqrt(deg_dst), deg = cnt + 1.
__global__ void __launch_bounds__(256)
gcn_scatter_kernel(const float* __restrict__ h, const int* __restrict__ src,
                   const int* __restrict__ dst, const float* __restrict__ cnt,
                   float* __restrict__ agg, int E) {
  int e = blockIdx.x * 8 + (threadIdx.x >> 5);
  if (e >= E) return;
  int lane = threadIdx.x & 31;
  int s = src[e], d = dst[e];
  float norm = rsqrtf(cnt[s] + 1.0f) * rsqrtf(cnt[d] + 1.0f);
  const float4 v = *(const float4*)(h + (size_t)s * C128 + lane * 4);
  float* ap = agg + (size_t)d * C128 + lane * 4;
  atomicAdd(ap + 0, v.x * norm);
  atomicAdd(ap + 1, v.y * norm);
  atomicAdd(ap + 2, v.z * norm);
  atomicAdd(ap + 3, v.w * norm);
}

// --------------------------------------------- fused SAGE GEMM (128 -> 128)
// out = relu( (agg/max(cnt,1)) @ Wl + x @ Wr + b ),  per-block 32x128 tile.
// 8 waves per block; wave w computes output columns [16w, 16w+16) for TWO
// 16-row tiles, so each B (weight) fragment load feeds 4 WMMAs.
__global__ void __launch_bounds__(256)
sage_gemm_kernel(const float* __restrict__ agg, const float* __restrict__ cnt,
                 const float* __restrict__ x, const float* __restrict__ Wl,
                 const float* __restrict__ Wr, const float* __restrict__ bias,
                 float* __restrict__ out, int N) {
  __shared__ float sMean[32 * C128];
  __shared__ float sX[32 * C128];
  const int m0 = blockIdx.x * 32;
  const int tid = threadIdx.x;

  // Stage both 32x128 A-tiles in LDS (mean computed on the fly).
  for (int i = tid; i < 32 * C128; i += 256) {
    int r = i >> 7, c = i & 127;
    int row = m0 + r;
    if (row >= N) row = N - 1;           // clamp (tail-safe)
    float inv = 1.0f / fmaxf(cnt[row], 1.0f);
    sMean[i] = agg[(size_t)row * C128 + c] * inv;
    sX[i]   = x[(size_t)row * C128 + c];
  }
  __syncthreads();

  const int wave = tid >> 5;
  const int lane = tid & 31;
  const int hh   = lane >> 4;            // half-wave select
  const int l16  = lane & 15;
  const int col  = wave * 16 + l16;      // output column for this lane

  v8f acc0 = {0.f, 0.f, 0.f, 0.f, 0.f, 0.f, 0.f, 0.f};
  v8f acc1 = {0.f, 0.f, 0.f, 0.f, 0.f, 0.f, 0.f, 0.f};
  #pragma unroll 4
  for (int kk = 0; kk < 32; ++kk) {
    const int k = kk * 4 + 2 * hh;       // K pair handled by this half-wave
    // B fragments: rows k, k+1 of W at this lane's column (shared by 4 WMMAs).
    v2f bl = { Wl[(size_t)k * C128 + col], Wl[(size_t)(k + 1) * C128 + col] };
    v2f br = { Wr[(size_t)k * C128 + col], Wr[(size_t)(k + 1) * C128 + col] };
    // A fragments, tile 0 (rows m0..m0+15) and tile 1 (rows m0+16..m0+31).
    v2f am0 = *(const v2f*)&sMean[l16 * C128 + k];
    v2f ax0 = *(const v2f*)&sX[l16 * C128 + k];
    v2f am1 = *(const v2f*)&sMean[(16 + l16) * C128 + k];
    v2f ax1 = *(const v2f*)&sX[(16 + l16) * C128 + k];
    acc0 = __builtin_amdgcn_wmma_f32_16x16x4_f32(false, am0, false, bl,
                                                 (short)0, acc0, false, false);
    acc0 = __builtin_amdgcn_wmma_f32_16x16x4_f32(false, ax0, false, br,
                                                 (short)0, acc0, false, false);
    acc1 = __builtin_amdgcn_wmma_f32_16x16x4_f32(false, am1, false, bl,
                                                 (short)0, acc1, false, false);
    acc1 = __builtin_amdgcn_wmma_f32_16x16x4_f32(false, ax1, false, br,
                                                 (short)0, acc1, false, false);
  }

  const float bv = bias[col];
  if (m0 + 32 <= N) {                    // fast path: unconditional stores
    #pragma unroll
    for (int i = 0; i < 8; ++i) {
      int r0 = m0 + i + 8 * hh;          // C/D layout: M = i + 8*half
      out[(size_t)r0 * C128 + col]        = fmaxf(acc0[i] + bv, 0.0f);
      out[(size_t)(r0 + 16) * C128 + col] = fmaxf(acc1[i] + bv, 0.0f);
    }
  } else {                               // tail block only
    #pragma unroll
    for (int i = 0; i < 8; ++i) {
      int r0 = m0 + i + 8 * hh;
      if (r0 < N)      out[(size_t)r0 * C128 + col]        = fmaxf(acc0[i] + bv, 0.0f);
      if (r0 + 16 < N) out[(size_t)(r0 + 16) * C128 + col] = fmaxf(acc1[i] + bv, 0.0f);
    }
  }
}

// ------------------------------------------------ plain GEMM for GCN (h @ W)
__global__ void __launch_bounds__(256)
gemm128_kernel(const float* __restrict__ x, const float* __restrict__ W,
               float* __restrict__ out, int N) {
  __shared__ float sX[32 * C128];
  const int m0 = blockIdx.x * 32;
  const int tid = threadIdx.x;
  for (int i = tid; i < 32 * C128; i += 256) {
    int r = i >> 7, c = i & 127;
    int row = m0 + r;
    if (row >= N) row = N - 1;
    sX[i] = x[(size_t)row * C128 + c];
  }
  __syncthreads();

  const int wave = tid >> 5;
  const int lane = tid & 31;
  const int hh   = lane >> 4;
  const int l16  = lane & 15;
  const int col  = wave * 16 + l16;

  v8f acc0 = {0.f, 0.f, 0.f, 0.f, 0.f, 0.f, 0.f, 0.f};
  v8f acc1 = {0.f, 0.f, 0.f, 0.f, 0.f, 0.f, 0.f, 0.f};
  #pragma unroll 4
  for (int kk = 0; kk < 32; ++kk) {
    const int k = kk * 4 + 2 * hh;
    v2f b = { W[(size_t)k * C128 + col], W[(size_t)(k + 1) * C128 + col] };
    v2f a0 = *(const v2f*)&sX[l16 * C128 + k];
    v2f a1 = *(const v2f*)&sX[(16 + l16) * C128 + k];
    acc0 = __builtin_amdgcn_wmma_f32_16x16x4_f32(false, a0, false, b,
                                                 (short)0, acc0, false, false);
    acc1 = __builtin_amdgcn_wmma_f32_16x16x4_f32(false, a1, false, b,
                                                 (short)0, acc1, false, false);
  }

  if (m0 + 32 <= N) {
    #pragma unroll
    for (int i = 0; i < 8; ++i) {
      int r0 = m0 + i + 8 * hh;
      out[(size_t)r0 * C128 + col]        = acc0[i];
      out[(size_t)(r0 + 16) * C128 + col] = acc1[i];
    }
  } else {
    #pragma unroll
    for (int i = 0; i < 8; ++i) {
      int r0 = m0 + i + 8 * hh;
      if (r0 < N)      out[(size_t)r0 * C128 + col]        = acc0[i];
      if (r0 + 16 < N) out[(size_t)(r0 + 16) * C128 + col] = acc1[i];
    }
  }
}

// ---------------------------------- GCN combine: relu(agg + t/deg + b)
__global__ void __launch_bounds__(256)
gcn_combine_kernel(const float* __restrict__ agg, const float* __restrict__ t,
                   const float* __restrict__ cnt, const float* __restrict__ bias,
                   float* __restrict__ out, int N) {
  size_t i = (size_t)blockIdx.x * blockDim.x + threadIdx.x;
  size_t total = (size_t)N * C128;
  if (i >= total) return;
  int row = (int)(i >> 7);
  int c = (int)(i & 127);
  float deg = cnt[row] + 1.0f;
  out[i] = fmaxf(agg[i] + t[i] / deg + bias[c], 0.0f);
}

// ------------------------------------------- final SAGE GEMM (128 -> 16)
// One wave per 16-row tile; writes pre-softmax logits [N,16].
__global__ void __launch_bounds__(32)
sage_out_gemm_kernel(const float* __restrict__ agg, const float* __restrict__ cnt,
                     const float* __restrict__ x, const float* __restrict__ Wl,
                     const float* __restrict__ Wr, const float* __restrict__ bias,
                     float* __restrict__ pre, int N) {
  const int m0 = blockIdx.x * 16;
  const int lane = threadIdx.x & 31;
  const int hh = lane >> 4;
  const int l16 = lane & 15;
  int arow = m0 + l16;
  if (arow >= N) arow = N - 1;
  const float inv = 1.0f / fmaxf(cnt[arow], 1.0f);
  const float* ar = agg + (size_t)arow * C128;
  const float* xr = x + (size_t)arow * C128;

  v8f acc = {0.f, 0.f, 0.f, 0.f, 0.f, 0.f, 0.f, 0.f};
  #pragma unroll 4
  for (int kk = 0; kk < 32; ++kk) {
    const int k = kk * 4 + 2 * hh;
    v2f am = { ar[k] * inv, ar[k + 1] * inv };
    v2f ax = { xr[k], xr[k + 1] };
    v2f bl = { Wl[(size_t)k * C16 + l16], Wl[(size_t)(k + 1) * C16 + l16] };
    v2f br = { Wr[(size_t)k * C16 + l16], Wr[(size_t)(k + 1) * C16 + l16] };
    acc = __builtin_amdgcn_wmma_f32_16x16x4_f32(false, am, false, bl,
                                                (short)0, acc, false, false);
    acc = __builtin_amdgcn_wmma_f32_16x16x4_f32(false, ax, false, br,
                                                (short)0, acc, false, false);
  }
  const float bv = bias[l16];
  if (m0 + 16 <= N) {
    #pragma unroll
    for (int i = 0; i < 8; ++i) {
      int row = m0 + i + 8 * hh;
      pre[(size_t)row * C16 + l16] = acc[i] + bv;
    }
  } else {
    #pragma unroll
    for (int i = 0; i < 8; ++i) {
      int row = m0 + i + 8 * hh;
      if (row < N) pre[(size_t)row * C16 + l16] = acc[i] + bv;
    }
  }
}

// ------------------------------------------------- log_softmax over 16 cols
__global__ void __launch_bounds__(256)
logsoftmax16_kernel(const float* __restrict__ pre, float* __restrict__ out, int N) {
  int row = blockIdx.x * blockDim.x + threadIdx.x;
  if (row >= N) return;
  const float* p = pre + (size_t)row * C16;
  float v[16];
  float m = -3.402823466e+38f;
  #pragma unroll
  for (int i = 0; i < 16; ++i) { v[i] = p[i]; m = fmaxf(m, v[i]); }
  float s = 0.0f;
  #pragma unroll
  for (int i = 0; i < 16; ++i) s += expf(v[i] - m);
  const float l = logf(s);
  float* o = out + (size_t)row * C16;
  #pragma unroll
  for (int i = 0; i < 16; ++i) o[i] = v[i] - m - l;
}

// ---------------------------------------------------------------- dispatcher

extern "C" void kernel_launch(void* const* d_in, const int* in_sizes, int n_in,
                              void* d_out, int out_size, void* d_ws, size_t ws_size,
                              hipStream_t stream) {
  const float* x    = (const float*)d_in[0];
  const int*   ei   = (const int*)d_in[1];
  const float* s0Wl = (const float*)d_in[2];
  const float* s0Wr = (const float*)d_in[3];
  const float* s0b  = (const float*)d_in[4];
  const float* g1W  = (const float*)d_in[5];
  const float* g1b  = (const float*)d_in[6];
  const float* s2Wl = (const float*)d_in[7];
  const float* s2Wr = (const float*)d_in[8];
  const float* s2b  = (const float*)d_in[9];
  const float* s3Wl = (const float*)d_in[10];
  const float* s3Wr = (const float*)d_in[11];
  const float* s3b  = (const float*)d_in[12];

  const int N = in_sizes[0] / C128;
  const int E = in_sizes[1] / 2;
  const int* src = ei;
  const int* dst = ei + E;

  // Workspace layout: agg[N*128] | cnt[N (padded to 64)] | hA[N*128] | hB[N*128]
  float* agg = (float*)d_ws;
  float* cnt = agg + (size_t)N * C128;
  float* hA  = cnt + (((size_t)N + 63) & ~(size_t)63);
  float* hB  = hA + (size_t)N * C128;

  const size_t nh = (size_t)N * C128;
  const int zgrid = 2048;
  const int egrid = (E + 7) / 8;
  const int g32   = (N + 31) / 32;
  const int g16   = (N + 15) / 16;
  const int cgrid = (int)((nh + 255) / 256);
  const int rgrid = (N + 255) / 256;

  // ---- Layer 0: SAGE(128->128) + ReLU  -> hA
  zero_kernel<<<zgrid, 256, 0, stream>>>(agg, nh + (size_t)N);  // agg + cnt
  scatter_sum_kernel<<<egrid, 256, 0, stream>>>(x, src, dst, agg, cnt, E);
  sage_gemm_kernel<<<g32, 256, 0, stream>>>(agg, cnt, x, s0Wl, s0Wr, s0b, hA, N);

  // ---- Layer 1: GCN(128->128) + ReLU  -> hA  (t = hB)
  gemm128_kernel<<<g32, 256, 0, stream>>>(hA, g1W, hB, N);
  zero_kernel<<<zgrid, 256, 0, stream>>>(agg, nh);
  gcn_scatter_kernel<<<egrid, 256, 0, stream>>>(hB, src, dst, cnt, agg, E);
  gcn_combine_kernel<<<cgrid, 256, 0, stream>>>(agg, hB, cnt, g1b, hA, N);

  // ---- Layer 2: SAGE(128->128) + ReLU  -> hB
  zero_kernel<<<zgrid, 256, 0, stream>>>(agg, nh);
  scatter_sum_kernel<<<egrid, 256, 0, stream>>>(hA, src, dst, agg, nullptr, E);
  sage_gemm_kernel<<<g32, 256, 0, stream>>>(agg, cnt, hA, s2Wl, s2Wr, s2b, hB, N);

  // ---- Layer 3: SAGE(128->16), no ReLU -> hA (pre-logits), then log_softmax
  zero_kernel<<<zgrid, 256, 0, stream>>>(agg, nh);
  scatter_sum_kernel<<<egrid, 256, 0, stream>>>(hB, src, dst, agg, nullptr, E);
  sage_out_gemm_kernel<<<g16, 32, 0, stream>>>(agg, cnt, hB, s3Wl, s3Wr, s3b,
                                               hA, N);
  logsoftmax16_kernel<<<rgrid, 256, 0, stream>>>(hA, (float*)d_out, N);
}